// GCN_21174188770104
// MI455X (gfx1250) — compile-verified
//
#include <hip/hip_runtime.h>

typedef __attribute__((ext_vector_type(2))) float v2f;
typedef __attribute__((ext_vector_type(8))) float v8f;
typedef __attribute__((ext_vector_type(4))) int   v4i;
typedef __attribute__((address_space(1))) v4i     gv4i;  // global <4 x i32>
typedef __attribute__((address_space(3))) v4i     lv4i;  // LDS    <4 x i32>

#if (__has_builtin(__builtin_amdgcn_global_load_async_to_lds_b128) && \
     __has_builtin(__builtin_amdgcn_s_wait_asynccnt))
#define HAS_ASYNC_LDS 1
#else
#define HAS_ASYNC_LDS 0
#endif

// ---------------------------------------------------------------------------
// Degree / normalization kernels
// ---------------------------------------------------------------------------
__global__ void deg_init_k(float* __restrict__ deg, int n) {
    int i = blockIdx.x * blockDim.x + threadIdx.x;
    if (i < n) deg[i] = 1.0f;  // self-loop contributes 1 to every node's degree
}

__global__ void deg_count_k(float* __restrict__ deg,
                            const long long* __restrict__ ei, int E) {
    int e = blockIdx.x * blockDim.x + threadIdx.x;
    if (e < E) {
        int d = (int)ei[(size_t)E + e];  // dst row
        atomicAdd(&deg[d], 1.0f);
    }
}

__global__ void dinv_k(float* __restrict__ deg, int n) {
    int i = blockIdx.x * blockDim.x + threadIdx.x;
    if (i < n) deg[i] = rsqrtf(deg[i]);  // deg >= 1 always (self loop)
}

// ---------------------------------------------------------------------------
// Pack B[K,N] into per-wave WMMA fragment order:
//   Bp[((ct*K4 + k4)*32 + lane)*2 + j] = B[(4*k4 + 2*(lane>>4) + j)*N + ct*16 + (lane&15)]
// so each wave's per-iteration B fragment is one contiguous 256B chunk and each
// lane issues a single coalesced global_load_b64.
// ---------------------------------------------------------------------------
__global__ void pack_b_k(const float* __restrict__ B, float* __restrict__ Bp,
                         int N, int K) {
    int idx = blockIdx.x * blockDim.x + threadIdx.x;
    int total = N * K;
    if (idx >= total) return;
    int K4     = K >> 2;
    int per_ct = K4 * 64;
    int ct  = idx / per_ct;
    int rem = idx - ct * per_ct;
    int k4  = rem >> 6;
    int t   = rem & 63;
    int lane = t >> 1;
    int j    = t & 1;
    int half = lane >> 4;
    int mr   = lane & 15;
    Bp[idx] = B[(size_t)(4 * k4 + 2 * half + j) * N + ct * 16 + mr];
}

// ---------------------------------------------------------------------------
// WMMA f32 GEMM: C[M,N] = A[M,K] @ Bp(packed) (+ bias)
// Block = 256 threads = 8 waves. ctiles = N/16 column tiles; each block covers
// rtiles = 8/ctiles row-groups of 16 rows. A tile staged in LDS (async direct-
// to-LDS when available) and reused by all waves of the block.
// ---------------------------------------------------------------------------
__global__ __launch_bounds__(256)
void wmma_gemm_k(const float* __restrict__ A, const float* __restrict__ Bp,
                 const float* __restrict__ bias, float* __restrict__ C,
                 int M, int N, int K, int ctiles) {
    extern __shared__ float lds[];
    const int rtiles = 8 / ctiles;
    const int tid    = threadIdx.x;
    const int mbase  = blockIdx.x * rtiles * 16;
    const int K4     = K >> 2;

    // Cooperative stage of the A tile (rtiles*16 rows x K cols) into LDS.
    const float4* A4 = (const float4*)A;
    float4*       L4 = (float4*)lds;
    const int cnt4 = rtiles * 16 * K4;
#if HAS_ASYNC_LDS
    for (int i = tid; i < cnt4; i += 256) {
        int row = i / K4;
        int c4  = i - row * K4;
        int g   = mbase + row;
        if (g < M) {
            __builtin_amdgcn_global_load_async_to_lds_b128(
                (gv4i*)(A4 + (size_t)g * K4 + c4),
                (lv4i*)(L4 + i),
                0, 0);
        } else {
            L4[i] = make_float4(0.f, 0.f, 0.f, 0.f);
        }
    }
    __builtin_amdgcn_s_wait_asynccnt(0);   // our async copies have landed in LDS
    __syncthreads();                        // all waves' copies visible
#else
    for (int i = tid; i < cnt4; i += 256) {
        int row = i / K4;
        int c4  = i - row * K4;
        int g   = mbase + row;
        float4 v = make_float4(0.f, 0.f, 0.f, 0.f);
        if (g < M) v = A4[(size_t)g * K4 + c4];
        L4[i] = v;
    }
    __syncthreads();
#endif

    const int wave = tid >> 5;
    const int lane = tid & 31;
    const int half = lane >> 4;   // 0: lanes 0-15, 1: lanes 16-31
    const int mr   = lane & 15;
    const int rt   = wave / ctiles;
    const int ct   = wave - rt * ctiles;
    const int m0   = mbase + rt * 16;
    if (m0 >= M) return;          // wave-uniform; EXEC stays all-1s for WMMA
    const int n0   = ct * 16;

    // A frag: lane half selects K offset {0,1} vs {2,3}; row = rt*16 + mr.
    const float* ap = &lds[(size_t)(rt * 16 + mr) * K + 2 * half];
    // Packed B frag: one v2f per lane per iteration, contiguous per wave.
    const v2f* bp = (const v2f*)Bp + (size_t)ct * K4 * 32 + lane;

    v8f acc = {};
    for (int k = 0; k < K; k += 4) {
        v2f a = *(const v2f*)ap;   // ds_load_b64 (pairs merge to ds_load_2addr_b64)
        v2f b = *bp;               // coalesced global_load_b64
        acc = __builtin_amdgcn_wmma_f32_16x16x4_f32(
            /*neg_a=*/false, a, /*neg_b=*/false, b,
            /*c_mod=*/(short)0, acc, /*reuse_a=*/false, /*reuse_b=*/false);
        ap += 4;
        bp += 32;
    }

    const float bv = bias ? bias[n0 + mr] : 0.0f;
#pragma unroll
    for (int r = 0; r < 8; ++r) {
        int row = m0 + r + 8 * half;      // C/D layout: VGPR r -> M = r + 8*half
        C[(size_t)row * N + n0 + mr] = acc[r] + bv;
    }
}

// ---------------------------------------------------------------------------
// Aggregation: out[i] = h[i]*dinv[i]^2 (self loop), then edge scatter-add,
// then +bias and ReLU.  F is fixed at 128 (32 lanes x float4).
// ---------------------------------------------------------------------------
__global__ void self_loop_k(const float* __restrict__ h,
                            const float* __restrict__ dinv,
                            float* __restrict__ out, int n) {
    int idx  = blockIdx.x * blockDim.x + threadIdx.x;
    int node = idx >> 5;
    int lane = idx & 31;
    if (node >= n) return;
    float w = dinv[node];
    w = w * w;
    float4 v = ((const float4*)(h + (size_t)node * 128))[lane];
    v.x *= w; v.y *= w; v.z *= w; v.w *= w;
    ((float4*)(out + (size_t)node * 128))[lane] = v;
}

__global__ void edge_scatter_k(const float* __restrict__ h,
                               const long long* __restrict__ ei,
                               const float* __restrict__ dinv,
                               float* __restrict__ out, int E) {
    int e    = blockIdx.x * (blockDim.x >> 5) + (threadIdx.x >> 5);
    int lane = threadIdx.x & 31;
    if (e >= E) return;
    int s = (int)ei[e];
    int d = (int)ei[(size_t)E + e];
    float norm = dinv[s] * dinv[d];
    float4 v = ((const float4*)(h + (size_t)s * 128))[lane];  // 512B coalesced gather
    float* o = out + (size_t)d * 128 + lane * 4;
    atomicAdd(o + 0, v.x * norm);   // non-returning global_atomic_add_f32, L2-resident
    atomicAdd(o + 1, v.y * norm);
    atomicAdd(o + 2, v.z * norm);
    atomicAdd(o + 3, v.w * norm);
}

__global__ void bias_relu_k(float* __restrict__ x, const float* __restrict__ b, int n) {
    int idx  = blockIdx.x * blockDim.x + threadIdx.x;
    int node = idx >> 5;
    int lane = idx & 31;
    if (node >= n) return;
    float4 v  = ((float4*)(x + (size_t)node * 128))[lane];
    float4 bb = ((const float4*)b)[lane];
    v.x = fmaxf(v.x + bb.x, 0.f);
    v.y = fmaxf(v.y + bb.y, 0.f);
    v.z = fmaxf(v.z + bb.z, 0.f);
    v.w = fmaxf(v.w + bb.w, 0.f);
    ((float4*)(x + (size_t)node * 128))[lane] = v;
}

// ---------------------------------------------------------------------------
extern "C" void kernel_launch(void* const* d_in, const int* in_sizes, int n_in,
                              void* d_out, int out_size, void* d_ws, size_t ws_size,
                              hipStream_t stream) {
    const float*     x  = (const float*)d_in[0];
    const long long* ei = (const long long*)d_in[1];   // int64 [2, E]
    const float*     W1 = (const float*)d_in[2];
    const float*     b1 = (const float*)d_in[3];
    const float*     W2 = (const float*)d_in[4];
    const float*     b2 = (const float*)d_in[5];
    const float*     Wo = (const float*)d_in[6];
    const float*     bo = (const float*)d_in[7];
    float*           out = (float*)d_out;

    const int NNODES = 100000;
    const int E      = in_sizes[1] / 2;        // 1,600,000
    const int FIN    = in_sizes[0] / NNODES;   // 512
    const int EMB    = in_sizes[3];            // 128
    const int NCLS   = in_sizes[7];            // 16

    // Workspace layout: bufH | bufG | dinv | Bp1 | Bp2 | Bp3
    float* bufH = (float*)d_ws;
    float* bufG = bufH + (size_t)NNODES * EMB;
    float* dinv = bufG + (size_t)NNODES * EMB;
    float* Bp1  = dinv + NNODES;               // FIN*EMB   (64K floats)
    float* Bp2  = Bp1 + (size_t)FIN * EMB;     // EMB*EMB   (16K floats)
    float* Bp3  = Bp2 + (size_t)EMB * EMB;     // EMB*NCLS  (2K floats)

    // ---- degrees + rsqrt normalization (shared by both conv layers) ----
    deg_init_k <<<(NNODES + 255) / 256, 256, 0, stream>>>(dinv, NNODES);
    deg_count_k<<<(E + 255) / 256,      256, 0, stream>>>(dinv, ei, E);
    dinv_k     <<<(NNODES + 255) / 256, 256, 0, stream>>>(dinv, NNODES);

    // ---- pack weights into WMMA fragment order (once per launch) ----
    pack_b_k<<<(FIN * EMB + 255) / 256,  256, 0, stream>>>(W1, Bp1, EMB,  FIN);
    pack_b_k<<<(EMB * EMB + 255) / 256,  256, 0, stream>>>(W2, Bp2, EMB,  EMB);
    pack_b_k<<<(EMB * NCLS + 255) / 256, 256, 0, stream>>>(Wo, Bp3, NCLS, EMB);

    const int rowTiles = (NNODES + 15) / 16;   // 6250

    // ---- layer 1: h = x @ W1 ; aggregate ; +b1 ; relu ----
    {
        int ctiles = EMB / 16;                           // 8
        int rtiles = 8 / ctiles;                         // 1
        int blocks = (rowTiles + rtiles - 1) / rtiles;   // 6250
        size_t ldsB = (size_t)rtiles * 16 * FIN * sizeof(float);  // 32 KB
        wmma_gemm_k<<<blocks, 256, ldsB, stream>>>(x, Bp1, nullptr, bufH,
                                                   NNODES, EMB, FIN, ctiles);
    }
    self_loop_k  <<<(NNODES * 32 + 255) / 256, 256, 0, stream>>>(bufH, dinv, bufG, NNODES);
    edge_scatter_k<<<(E + 7) / 8,              256, 0, stream>>>(bufH, ei, dinv, bufG, E);
    bias_relu_k  <<<(NNODES * 32 + 255) / 256, 256, 0, stream>>>(bufG, b1, NNODES);

    // ---- layer 2: h = g @ W2 ; aggregate ; +b2 ; relu ----
    {
        int ctiles = EMB / 16;                           // 8
        int rtiles = 8 / ctiles;                         // 1
        int blocks = (rowTiles + rtiles - 1) / rtiles;   // 6250
        size_t ldsB = (size_t)rtiles * 16 * EMB * sizeof(float);  // 8 KB
        wmma_gemm_k<<<blocks, 256, ldsB, stream>>>(bufG, Bp2, nullptr, bufH,
                                                   NNODES, EMB, EMB, ctiles);
    }
    self_loop_k  <<<(NNODES * 32 + 255) / 256, 256, 0, stream>>>(bufH, dinv, bufG, NNODES);
    edge_scatter_k<<<(E + 7) / 8,              256, 0, stream>>>(bufH, ei, dinv, bufG, E);
    bias_relu_k  <<<(NNODES * 32 + 255) / 256, 256, 0, stream>>>(bufG, b2, NNODES);

    // ---- classifier: out = g @ Wo + bo ----
    {
        int ctiles = NCLS / 16;                          // 1
        int rtiles = 8 / ctiles;                         // 8
        int blocks = (rowTiles + rtiles - 1) / rtiles;   // 782 (last block guarded)
        size_t ldsB = (size_t)rtiles * 16 * EMB * sizeof(float);  // 64 KB
        wmma_gemm_k<<<blocks, 256, ldsB, stream>>>(bufG, Bp3, bo, out,
                                                   NNODES, NCLS, EMB, ctiles);
    }
}